// LengthRegulator_86294482911718
// MI455X (gfx1250) — compile-verified
//
#include <hip/hip_runtime.h>
#include <stdint.h>

#define B_SZ    32
#define T_INLEN 512
#define D_SZ    512
#define MAXLEN  4096

typedef __attribute__((ext_vector_type(4))) float v4f;

// ---------------------------------------------------------------------------
// Kernel 1: per-batch inclusive scan of durations (512 elems) + mel_length.
// 32 blocks x 512 threads, Hillis-Steele in LDS. Tiny vs the 256MB write.
// ---------------------------------------------------------------------------
__global__ __launch_bounds__(512) void lr_cumsum_kernel(const int* __restrict__ dur,
                                                        int* __restrict__ cum,
                                                        int* __restrict__ mel_out) {
  __shared__ int s[T_INLEN];
  const int b = blockIdx.x;
  const int t = threadIdx.x;
  s[t] = dur[b * T_INLEN + t];
  __syncthreads();
#pragma unroll
  for (int off = 1; off < T_INLEN; off <<= 1) {
    const int add = (t >= off) ? s[t - off] : 0;
    __syncthreads();
    s[t] += add;
    __syncthreads();
  }
  cum[b * T_INLEN + t] = s[t];
  if (t == T_INLEN - 1) mel_out[b] = s[t];   // int32 bits into f32 output tail
}

// ---------------------------------------------------------------------------
// Kernel 2: expand. One block = one (batch, 32-frame tile). 256 thr = 8 waves.
//  - cum[b,:] (2KB) staged into LDS via gfx1250 async global->LDS B128 copies
//    (ASYNCcnt path), so searches run out of LDS with no VGPR round trip.
//  - each wave: 4 frames; uniform binary search; 4x float4 coalesced row copy
//    with NON-TEMPORAL stores (keep L2 for x, stream the 256MB output).
// ---------------------------------------------------------------------------
__global__ __launch_bounds__(256) void lr_expand_kernel(const float* __restrict__ x,
                                                        const int* __restrict__ cum,
                                                        float* __restrict__ out) {
  __shared__ __align__(16) int s_cum[T_INLEN];

  const int b    = blockIdx.x >> 7;       // 128 tiles per batch
  const int tile = blockIdx.x & 127;
  const int t0   = tile * 32;
  const int tid  = threadIdx.x;

  // Async stage: 128 lanes x 16B = 2048B = cum[b, 0..511]
  if (tid < T_INLEN / 4) {
    const int* gsrc = cum + b * T_INLEN + tid * 4;
    // Flat LDS address: low 32 bits are the wave-relative LDS byte offset.
    unsigned lds_off = (unsigned)(uintptr_t)(&s_cum[0]) + (unsigned)tid * 16u;
    asm volatile("global_load_async_to_lds_b128 %0, %1, off"
                 :: "v"(lds_off), "v"(gsrc)
                 : "memory");
  }
  asm volatile("s_wait_asynccnt 0x0" ::: "memory");
  __syncthreads();

  const int wave = tid >> 5;
  const int lane = tid & 31;
  const int mel  = s_cum[T_INLEN - 1];

  const v4f* __restrict__ x4 = (const v4f*)(x + (size_t)b * T_INLEN * D_SZ);
  v4f* __restrict__ o4       = (v4f*)(out + ((size_t)b * MAXLEN + t0) * D_SZ);

  for (int f = wave; f < 32; f += 8) {
    const int t = t0 + f;
    v4f* __restrict__ orow = o4 + (size_t)f * (D_SZ / 4);
    if (t < mel) {
      // upper_bound(cum, t): count of cum[i] <= t  (wave-uniform search)
      int lo = 0, hi = T_INLEN;
      while (lo < hi) {
        const int mid = (lo + hi) >> 1;
        if (s_cum[mid] <= t) lo = mid + 1; else hi = mid;
      }
      const int idx = lo > (T_INLEN - 1) ? (T_INLEN - 1) : lo;
      const v4f* __restrict__ xr = x4 + (size_t)idx * (D_SZ / 4);
#pragma unroll
      for (int k = 0; k < 4; ++k) {
        const v4f v = xr[lane + 32 * k];                       // L2-resident
        __builtin_nontemporal_store(v, &orow[lane + 32 * k]);  // stream out
      }
    } else {
      const v4f z = {0.0f, 0.0f, 0.0f, 0.0f};
#pragma unroll
      for (int k = 0; k < 4; ++k)
        __builtin_nontemporal_store(z, &orow[lane + 32 * k]);
    }
  }
}

// ---------------------------------------------------------------------------
extern "C" void kernel_launch(void* const* d_in, const int* in_sizes, int n_in,
                              void* d_out, int out_size, void* d_ws, size_t ws_size,
                              hipStream_t stream) {
  const float* x   = (const float*)d_in[0];   // (32, 512, 512) f32
  const int*   dur = (const int*)d_in[1];     // (32, 512) i32
  // d_in[2] = max_length scalar; fixed at 4096 per reference.

  float* out = (float*)d_out;                                   // (32,4096,512)
  int*   mel = (int*)(out + (size_t)B_SZ * MAXLEN * D_SZ);      // (32,) i32 bits
  int*   cum = (int*)d_ws;                                      // 64 KB scratch

  lr_cumsum_kernel<<<B_SZ, T_INLEN, 0, stream>>>(dur, cum, mel);
  lr_expand_kernel<<<B_SZ * (MAXLEN / 32), 256, 0, stream>>>(x, cum, out);
}